// Converse2D_Up_80771154968994
// MI455X (gfx1250) — compile-verified
//
#include <hip/hip_runtime.h>
#include <math.h>

// ---------------------------------------------------------------------------
// Converse2D (scale=2, kernel=3, wrap pad=2) for MI455X / gfx1250.
// DFT-as-GEMM on V_WMMA_F32_16X16X4_F32 (f32 needed: spectral solve divides
// by biaseps ~ 1.33e-4, amplifying error ~7.5e3x).
// Working planes padded 264 -> 272 so every WMMA tile is full; DFT-matrix pad
// is zero so pad columns contribute nothing.
// Block = 8 waves sharing one n-tile: the 264x16 complex DFT panel (33.8 KB)
// is staged once per block into LDS via GLOBAL_LOAD_ASYNC_TO_LDS_B128
// (ASYNCcnt), so the K loop issues a single global b64 (A operand) per step.
// Workspace requirement: ~665 MB (2244 planes of 272*272 f32).
// ---------------------------------------------------------------------------

constexpr int B    = 4;
constexpr int C    = 64;
constexpr int HIN  = 128;
constexpr int S    = 2;
constexpr int PADW = 2;          // wrap pad
constexpr int HP   = 132;        // padded spatial (128 + 2*2)
constexpr int N    = 264;        // FFT size = HP * S
constexpr int NP   = 272;        // padded to multiple of 16
constexpr int PLANE = NP * NP;   // 73984 floats per plane
constexpr int HOUT = 256;        // cropped output size

constexpr int MROWS  = B * C * N;        // 67584 vectors per 1-D pass
constexpr int MTILES = MROWS / 16;       // 4224
constexpr int NTILES = NP / 16;          // 17
constexpr int WPB    = 8;                // waves (= m-tiles) per 256-thread block
constexpr int GEMM_GRID = NTILES * (MTILES / WPB);   // 17 * 528 = 8976

constexpr float PI = 3.14159265358979323846f;

typedef float v2f __attribute__((ext_vector_type(2)));
typedef float v8f __attribute__((ext_vector_type(8)));
typedef int   v4i __attribute__((ext_vector_type(4)));

#if __has_builtin(__builtin_amdgcn_global_load_async_to_lds_b128)
#define HAVE_ASYNC_LDS 1
typedef __attribute__((address_space(1))) v4i* gv4i_p;   // global src
typedef __attribute__((address_space(3))) v4i* lv4i_p;   // LDS dst
#else
#define HAVE_ASYNC_LDS 0
#endif

// workspace layout (float offsets)
constexpr long OFF_DFTF_R = 0;
constexpr long OFF_DFTF_I = OFF_DFTF_R + PLANE;
constexpr long OFF_DFTI_R = OFF_DFTF_I + PLANE;
constexpr long OFF_DFTI_I = OFF_DFTI_R + PLANE;
constexpr long OFF_FB_R   = OFF_DFTI_I + PLANE;
constexpr long OFF_FB_I   = OFF_FB_R + (long)C * PLANE;
constexpr long OFF_F2B    = OFF_FB_I + (long)C * PLANE;
constexpr long OFF_RA     = OFF_F2B + (long)C * PLANE;           // real tensor / final spatial
constexpr long OFF_RB     = OFF_RA + (long)B * C * PLANE;
constexpr long OFF_TMP_R  = OFF_RB + (long)B * C * PLANE;
constexpr long OFF_TMP_I  = OFF_TMP_R + (long)B * C * PLANE;
constexpr long OFF_YA_R   = OFF_TMP_I + (long)B * C * PLANE;
constexpr long OFF_YA_I   = OFF_YA_R + (long)B * C * PLANE;
constexpr long OFF_YB_R   = OFF_YA_I + (long)B * C * PLANE;
constexpr long OFF_YB_I   = OFF_YB_R + (long)B * C * PLANE;

__device__ __forceinline__ float biaseps(float bb) {
    return 1.0f / (1.0f + expf(9.0f - bb)) + 1e-5f;   // sigmoid(b-9)+eps
}

// Stage 16 bytes global -> LDS. CDNA5 path: async copy tracked by ASYNCcnt,
// no VGPR round trip. Fallback: plain vector copy through VGPRs.
__device__ __forceinline__ void stage16(const float* __restrict__ g, float* l) {
#if HAVE_ASYNC_LDS
    __builtin_amdgcn_global_load_async_to_lds_b128((gv4i_p)g, (lv4i_p)l, 0, 0);
#else
    *(float4*)l = *(const float4*)g;
#endif
}

__device__ __forceinline__ void stage_wait() {
#if HAVE_ASYNC_LDS
#if __has_builtin(__builtin_amdgcn_s_wait_asynccnt)
    __builtin_amdgcn_s_wait_asynccnt(0);
#else
    asm volatile("s_wait_asynccnt 0" ::: "memory");
#endif
#endif
}

// ---------------------------------------------------------------------------
// DFT tables: forward e^{-2pi i kn/N}; inverse e^{+2pi i kn/N}/N (1/N folded
// into each inverse pass -> 1/N^2 total). Pad region zeroed.
// ---------------------------------------------------------------------------
__global__ void build_dft_tables(float* __restrict__ fr, float* __restrict__ fi,
                                 float* __restrict__ ir_, float* __restrict__ ii_) {
    int idx = blockIdx.x * blockDim.x + threadIdx.x;
    if (idx >= PLANE) return;
    int k = idx / NP, n = idx % NP;
    float vr = 0.f, vi = 0.f, wr = 0.f, wi = 0.f;
    if (k < N && n < N) {
        int t = (k * n) % N;                      // exact integer phase reduction
        float ang = -2.0f * PI * (float)t / (float)N;
        vr = cosf(ang); vi = sinf(ang);
        wr = vr * (1.0f / (float)N);
        wi = -vi * (1.0f / (float)N);
    }
    fr[idx] = vr; fi[idx] = vi; ir_[idx] = wr; ii_[idx] = wi;
}

// ---------------------------------------------------------------------------
// OTF of rolled 3x3 PSF: FB[u,v] = sum_{i,j} w[i,j] e^{-2pi i (u(i-1)+v(j-1))/N}
// ---------------------------------------------------------------------------
__global__ void build_otf(const float* __restrict__ wgt,
                          float* __restrict__ FBr, float* __restrict__ FBi,
                          float* __restrict__ F2B) {
    long idx = (long)blockIdx.x * blockDim.x + threadIdx.x;
    if (idx >= (long)C * PLANE) return;
    int p = (int)(idx % PLANE); int c = (int)(idx / PLANE);
    int u = p / NP, v = p % NP;
    float fr = 0.f, fi = 0.f;
    if (u < N && v < N) {
#pragma unroll
        for (int i = 0; i < 3; ++i)
#pragma unroll
            for (int j = 0; j < 3; ++j) {
                float wv = wgt[(c * 3 + i) * 3 + j];
                int ph = (u * (i - 1) + v * (j - 1)) % N;
                if (ph < 0) ph += N;
                float ang = -2.0f * PI * (float)ph / (float)N;
                fr += wv * cosf(ang);
                fi += wv * sinf(ang);
            }
    }
    FBr[idx] = fr; FBi[idx] = fi; F2B[idx] = fr * fr + fi * fi;
}

// ---------------------------------------------------------------------------
// Fused wrap-pad + upsample: RA = zero-upsample(xp), RB = biaseps * nearest-up.
// Writes whole padded plane (pads zeroed; ws is poisoned by the harness).
// ---------------------------------------------------------------------------
__global__ void build_spatial(const float* __restrict__ x, const float* __restrict__ bias,
                              float* __restrict__ RA, float* __restrict__ RB) {
    long idx = (long)blockIdx.x * blockDim.x + threadIdx.x;
    if (idx >= (long)B * C * PLANE) return;
    int p = (int)(idx % PLANE); long t = idx / PLANE;
    int c = (int)(t % C); int b = (int)(t / C);
    int h = p / NP, w0 = p % NP;
    float sty = 0.f, xup = 0.f;
    if (h < N && w0 < N) {
        int hp = h >> 1, wp = w0 >> 1;          // padded coords [0,132)
        int hs = hp - PADW; hs += (hs < 0) ? HIN : 0; hs -= (hs >= HIN) ? HIN : 0;
        int ws_ = wp - PADW; ws_ += (ws_ < 0) ? HIN : 0; ws_ -= (ws_ >= HIN) ? HIN : 0;
        float v = x[(((long)b * C + c) * HIN + hs) * HIN + ws_];
        xup = biaseps(bias[c]) * v;
        sty = (((h | w0) & 1) == 0) ? v : 0.f;
    }
    RA[idx] = sty;
    RB[idx] = xup;
}

// ---------------------------------------------------------------------------
// Complex DFT pass as WMMA GEMM. Block = 8 waves x one shared n-tile; the
// D-panel (264x16, r+i) is staged to LDS once per block, then each wave runs
// its 16x16 tile with a single global b64 A-load per k-step.
//   row pass (COLPASS=0): X[m,k] at bc*PLANE + inner*NP + k
//   col pass (COLPASS=1): X[m,k] at bc*PLANE + k*NP + inner   (stride-NP K)
// REALIN skips the imaginary input GEMMs; REALOUT skips the imaginary store.
// ---------------------------------------------------------------------------
template <int COLPASS, int REALIN, int REALOUT>
__global__ __launch_bounds__(256) void dft_gemm(
    const float* __restrict__ Ar, const float* __restrict__ Ai,
    const float* __restrict__ Dr, const float* __restrict__ Di,
    float* __restrict__ Yr, float* __restrict__ Yi) {
    __shared__ __align__(16) float ldsBr[N * 16];   // D[k][n0+0..15] real
    __shared__ __align__(16) float ldsBi[N * 16];   // D[k][n0+0..15] imag

    const int bn  = blockIdx.x % NTILES;            // n-tile shared by all 8 waves
    const int bm0 = (blockIdx.x / NTILES) * WPB;    // first m-tile of this block
    const int n0  = bn * 16;

    // ---- stage the whole K-panel of D into LDS (one b128 per 4 floats) ----
    for (int idx = threadIdx.x; idx < N * 4; idx += 256) {
        const int k = idx >> 2, part = (idx & 3) * 4;
        stage16(Dr + (long)k * NP + n0 + part, ldsBr + k * 16 + part);
        stage16(Di + (long)k * NP + n0 + part, ldsBi + k * 16 + part);
    }
    stage_wait();
    __syncthreads();

    const int lane = threadIdx.x & 31;
    const int wv   = threadIdx.x >> 5;
    const int tm   = bm0 + wv;
    const int half = lane >> 4;                  // 0: K={0,1}, 1: K={2,3}
    const int l15  = lane & 15;

    // A operand: lane holds row (tm*16+l15), two consecutive K values
    const int rowIdx = tm * 16 + l15;
    const int bc     = rowIdx / N;
    const int inner  = rowIdx - bc * N;
    const long sk    = COLPASS ? NP : 1;
    const long baseA = COLPASS ? ((long)bc * PLANE + inner)
                               : ((long)bc * PLANE + (long)inner * NP);
    const int  kOff  = half * 2;
    const int  n     = n0 + l15;

    v8f arr = {}, ari = {}, aii = {}, air = {};
    for (int k0 = 0; k0 < N; k0 += 4) {
        const long ak = baseA + (long)(k0 + kOff) * sk;
        v2f a_r; a_r.x = Ar[ak]; a_r.y = Ar[ak + sk];
        const int bofs = (k0 + kOff) * 16 + l15;
        v2f b_r; b_r.x = ldsBr[bofs]; b_r.y = ldsBr[bofs + 16];
        v2f b_i; b_i.x = ldsBi[bofs]; b_i.y = ldsBi[bofs + 16];
        if (k0 + 8 < N) {                                       // gfx1250 global_prefetch_b8
            __builtin_prefetch(&Ar[baseA + (long)(k0 + 8 + kOff) * sk], 0, 0);
        }
        arr = __builtin_amdgcn_wmma_f32_16x16x4_f32(false, a_r, false, b_r, (short)0, arr, false, false);
        ari = __builtin_amdgcn_wmma_f32_16x16x4_f32(false, a_r, false, b_i, (short)0, ari, false, false);
        if constexpr (!REALIN) {
            v2f a_i; a_i.x = Ai[ak]; a_i.y = Ai[ak + sk];
            aii = __builtin_amdgcn_wmma_f32_16x16x4_f32(false, a_i, false, b_i, (short)0, aii, false, false);
            air = __builtin_amdgcn_wmma_f32_16x16x4_f32(false, a_i, false, b_r, (short)0, air, false, false);
        }
    }
    // C/D layout: VGPR v -> M = v + 8*half, N = lane&15
#pragma unroll
    for (int v = 0; v < 8; ++v) {
        const int mrow = tm * 16 + v + half * 8;
        const int obc  = mrow / N;
        const int oin  = mrow - obc * N;
        const long oaddr = COLPASS ? ((long)obc * PLANE + (long)n * NP + oin)
                                   : ((long)obc * PLANE + (long)oin * NP + n);
        float yr = arr[v];
        if constexpr (!REALIN) yr -= aii[v];
        Yr[oaddr] = yr;
        if constexpr (!REALOUT) {
            float yi = ari[v];
            if constexpr (!REALIN) yi += air[v];
            Yi[oaddr] = yi;
        }
    }
}

// ---------------------------------------------------------------------------
// Fused spectral solve. One thread owns the 4 quadrant-aliased frequencies
// (j+132du, l+132dv): FR = conj(FB)*Ya + Yb; FBR/invW = quadrant means of
// FB*FR and |FB|^2; invWBR = FBR/(invW+be); FX = (FR - conj(FB)*invWBR)/be.
// FX written in place over Ya.
// ---------------------------------------------------------------------------
__global__ void spectral_solve(const float* __restrict__ bias,
                               const float* __restrict__ FBr, const float* __restrict__ FBi,
                               const float* __restrict__ F2B,
                               float* __restrict__ Yar, float* __restrict__ Yai,
                               const float* __restrict__ Ybr, const float* __restrict__ Ybi) {
    long idx = (long)blockIdx.x * blockDim.x + threadIdx.x;
    if (idx >= (long)B * C * HP * HP) return;
    int l = (int)(idx % HP); long t = idx / HP;
    int j = (int)(t % HP); t /= HP;
    int c = (int)(t % C); int b = (int)(t / C);
    const long pb = (long)(b * C + c) * PLANE;
    const long fb = (long)c * PLANE;
    const float be = biaseps(bias[c]);

    long off[4];
    float frr[4], fri[4], fqr[4], fqi[4];
    float num_r = 0.f, num_i = 0.f, invw = 0.f;
#pragma unroll
    for (int q = 0; q < 4; ++q) {
        int du = q >> 1, dv = q & 1;
        off[q] = (long)(j + HP * du) * NP + (l + HP * dv);
        float br = FBr[fb + off[q]], bi = FBi[fb + off[q]];
        fqr[q] = br; fqi[q] = bi;
        float yar = Yar[pb + off[q]], yai = Yai[pb + off[q]];
        float ybr = Ybr[pb + off[q]], ybi = Ybi[pb + off[q]];
        float rr = br * yar + bi * yai + ybr;          // conj(FB)*Ya + Yb
        float ri = br * yai - bi * yar + ybi;
        frr[q] = rr; fri[q] = ri;
        num_r += br * rr - bi * ri;                    // FB * FR
        num_i += br * ri + bi * rr;
        invw  += F2B[fb + off[q]];
    }
    num_r *= 0.25f; num_i *= 0.25f; invw *= 0.25f;
    const float den = 1.0f / (invw + be);
    const float iwr = num_r * den, iwi = num_i * den;
    const float inv_be = 1.0f / be;
#pragma unroll
    for (int q = 0; q < 4; ++q) {
        float cr = fqr[q] * iwr + fqi[q] * iwi;        // conj(FB)*invWBR
        float ci = fqr[q] * iwi - fqi[q] * iwr;
        Yar[pb + off[q]] = (frr[q] - cr) * inv_be;
        Yai[pb + off[q]] = (fri[q] - ci) * inv_be;
    }
}

// ---------------------------------------------------------------------------
// Crop [4:260) and exact GELU.
// ---------------------------------------------------------------------------
__global__ void crop_gelu(const float* __restrict__ Z, float* __restrict__ out) {
    long idx = (long)blockIdx.x * blockDim.x + threadIdx.x;
    if (idx >= (long)B * C * HOUT * HOUT) return;
    int xc = (int)(idx % HOUT); long t = idx / HOUT;
    int yr = (int)(t % HOUT); t /= HOUT;   // t = b*C + c
    float v = Z[t * PLANE + (long)(yr + PADW * S) * NP + (xc + PADW * S)];
    out[idx] = 0.5f * v * (1.0f + erff(v * 0.70710678118654752f));
}

// ---------------------------------------------------------------------------
extern "C" void kernel_launch(void* const* d_in, const int* in_sizes, int n_in,
                              void* d_out, int out_size, void* d_ws, size_t ws_size,
                              hipStream_t stream) {
    const float* x    = (const float*)d_in[0];   // (4,64,128,128)
    const float* wgt  = (const float*)d_in[1];   // (1,64,3,3)
    const float* bias = (const float*)d_in[2];   // (1,64,1,1)
    float* out = (float*)d_out;
    float* ws  = (float*)d_ws;

    float* dftF_r = ws + OFF_DFTF_R; float* dftF_i = ws + OFF_DFTF_I;
    float* dftI_r = ws + OFF_DFTI_R; float* dftI_i = ws + OFF_DFTI_I;
    float* FBr = ws + OFF_FB_R; float* FBi = ws + OFF_FB_I; float* F2B = ws + OFF_F2B;
    float* RA  = ws + OFF_RA;   float* RB  = ws + OFF_RB;
    float* Tr  = ws + OFF_TMP_R; float* Ti = ws + OFF_TMP_I;
    float* Yar = ws + OFF_YA_R; float* Yai = ws + OFF_YA_I;
    float* Ybr = ws + OFF_YB_R; float* Ybi = ws + OFF_YB_I;

    const int TB = 256;
    build_dft_tables<<<(PLANE + TB - 1) / TB, TB, 0, stream>>>(dftF_r, dftF_i, dftI_r, dftI_i);
    build_otf<<<(int)(((long)C * PLANE + TB - 1) / TB), TB, 0, stream>>>(wgt, FBr, FBi, F2B);
    build_spatial<<<(int)(((long)B * C * PLANE + TB - 1) / TB), TB, 0, stream>>>(x, bias, RA, RB);

    // forward FFT of STy: real row pass -> complex col pass
    dft_gemm<0, 1, 0><<<GEMM_GRID, TB, 0, stream>>>(RA, nullptr, dftF_r, dftF_i, Tr, Ti);
    dft_gemm<1, 0, 0><<<GEMM_GRID, TB, 0, stream>>>(Tr, Ti, dftF_r, dftF_i, Yar, Yai);
    // forward FFT of biaseps * x_up
    dft_gemm<0, 1, 0><<<GEMM_GRID, TB, 0, stream>>>(RB, nullptr, dftF_r, dftF_i, Tr, Ti);
    dft_gemm<1, 0, 0><<<GEMM_GRID, TB, 0, stream>>>(Tr, Ti, dftF_r, dftF_i, Ybr, Ybi);

    // closed-form spectral solve (writes FX over Ya)
    spectral_solve<<<(int)(((long)B * C * HP * HP + TB - 1) / TB), TB, 0, stream>>>(
        bias, FBr, FBi, F2B, Yar, Yai, Ybr, Ybi);

    // inverse FFT (1/N per pass); final pass keeps only the real part
    dft_gemm<0, 0, 0><<<GEMM_GRID, TB, 0, stream>>>(Yar, Yai, dftI_r, dftI_i, Tr, Ti);
    dft_gemm<1, 0, 1><<<GEMM_GRID, TB, 0, stream>>>(Tr, Ti, dftI_r, dftI_i, RA, nullptr);

    crop_gelu<<<(int)(((long)B * C * HOUT * HOUT + TB - 1) / TB), TB, 0, stream>>>(RA, out);
}